// RetrievalLoss_26946624815701
// MI455X (gfx1250) — compile-verified
//
#include <hip/hip_runtime.h>

// out[i] = - sum_k Q[i,k] * Dm[i,k]   (N=131072 rows, D=256, fp32)
// HBM-bound: 256 MiB streamed once -> ~11.5 us floor at 23.3 TB/s.
//
// Per wave: one 16-row tile.
//   1) Stage Q-tile and Dm-tile (16x256 f32 each) into padded LDS using
//      GLOBAL_LOAD_ASYNC_TO_LDS_B128 (ASYNCcnt path): 64 loads issued
//      back-to-back with no VGPR round-trip -> ~32 KB in flight per wave,
//      absorbed by one s_wait_asynccnt 0. No __syncthreads needed (each
//      wave owns its LDS region).
//   2) C = Q_tile * Dm_tile^T via 64x V_WMMA_F32_16X16X4_F32 (ds_load_b64
//      fragment reads, row pad 260 -> all 64 banks hit exactly once).
//   3) Diagonal of C = the 16 row-dots; negate and store.

typedef __attribute__((ext_vector_type(2))) float v2f;
typedef __attribute__((ext_vector_type(8))) float v8f;

#define D_DIM           256
#define ROW_STRIDE      260   // 256 + 4 pad -> bank-conflict-free b64 reads
#define TILE_ROWS       16
#define WAVES_PER_BLOCK 4
#define BLOCK_THREADS   (WAVES_PER_BLOCK * 32)
#define TILE_F          (TILE_ROWS * ROW_STRIDE)   // floats per tile buffer

__global__ __launch_bounds__(BLOCK_THREADS, 1)
void RetrievalLoss_26946624815701_kernel(const float* __restrict__ q,
                                         const float* __restrict__ dm,
                                         float* __restrict__ out,
                                         int nrows)
{
    extern __shared__ float smem[];
    const int tid  = threadIdx.x;
    const int lane = tid & 31;          // wave32
    const int wv   = tid >> 5;
    const int tile = blockIdx.x * WAVES_PER_BLOCK + wv;
    const int row0 = tile * TILE_ROWS;

    float* qs = smem + wv * (2 * TILE_F);
    float* ds = qs + TILE_F;
    const bool active = (row0 + TILE_ROWS) <= nrows;
    if (!active) return;

    // ---- Async stage: global -> LDS, 128b per lane per op -------------
    // iteration it covers float4 elements i = lane + 32*it of the tile:
    //   row r = it>>1, col4 = lane + (it&1)*32
    //   global byte off = 512*it + lane*16           (row stride 1024 B)
    //   LDS    byte off = r*1040 + (it&1)*512 + lane*16 (row stride 1040 B)
    {
        const char* qbase = (const char*)(q  + (size_t)row0 * D_DIM);
        const char* dbase = (const char*)(dm + (size_t)row0 * D_DIM);
        unsigned goff  = (unsigned)lane * 16u;
        unsigned qloff = (unsigned)(uintptr_t)qs + (unsigned)lane * 16u;
        unsigned dloff = (unsigned)(uintptr_t)ds + (unsigned)lane * 16u;
#pragma unroll
        for (int it = 0; it < 32; ++it) {
            asm volatile("global_load_async_to_lds_b128 %0, %1, %2"
                         :: "v"(qloff), "v"(goff), "s"(qbase) : "memory");
            asm volatile("global_load_async_to_lds_b128 %0, %1, %2"
                         :: "v"(dloff), "v"(goff), "s"(dbase) : "memory");
            goff += 512u;
            const unsigned linc = (it & 1) ? 528u : 512u;
            qloff += linc;
            dloff += linc;
        }
        asm volatile("s_wait_asynccnt 0" ::: "memory");
    }

    // ---- WMMA f32 16x16x4: C += A(16x4) * B(4x16) ---------------------
    // Fragment addressing (identical for A and B in this diagonal GEMM):
    // lane l supplies a float2 from tile row (l%16), cols k0 + 2*(l>=16).
    const int r  = lane & 15;
    const int hk = (lane >> 4) << 1;     // 0 or 2
    const float* qrow = qs + r * ROW_STRIDE + hk;
    const float* drow = ds + r * ROW_STRIDE + hk;

    v8f c = {0.f, 0.f, 0.f, 0.f, 0.f, 0.f, 0.f, 0.f};
#pragma unroll 8
    for (int k0 = 0; k0 < D_DIM; k0 += 4) {
        v2f a = *(const v2f*)(qrow + k0);
        v2f b = *(const v2f*)(drow + k0);
        // (neg_a, A, neg_b, B, c_mod, C, reuse_a, reuse_b)
        c = __builtin_amdgcn_wmma_f32_16x16x4_f32(
                false, a, false, b, (short)0, c, false, false);
    }

    // ---- Diagonal extraction ------------------------------------------
    // C layout: VGPR v holds (M=v, N=lane) on lanes 0-15 and
    // (M=v+8, N=lane-16) on lanes 16-31.  (v,v) -> lane v;
    // (v+8,v+8) -> lane 24+v.
    float dval = 0.f;
#pragma unroll
    for (int v = 0; v < 8; ++v) {
        if (lane == v)      dval = c[v];
        if (lane == 24 + v) dval = c[v];
    }
    if (lane < 8)        out[row0 + lane]      = -dval;
    else if (lane >= 24) out[row0 + lane - 16] = -dval;
}

extern "C" void kernel_launch(void* const* d_in, const int* in_sizes, int n_in,
                              void* d_out, int out_size, void* d_ws, size_t ws_size,
                              hipStream_t stream) {
    (void)n_in; (void)d_ws; (void)ws_size; (void)out_size;
    const float* q  = (const float*)d_in[0];
    const float* dm = (const float*)d_in[1];
    float* out = (float*)d_out;

    const int nrows  = in_sizes[0] / D_DIM;                 // 131072
    const int tiles  = nrows / TILE_ROWS;                   // 8192
    const int blocks = (tiles + WAVES_PER_BLOCK - 1) / WAVES_PER_BLOCK;
    const size_t shmem = (size_t)WAVES_PER_BLOCK * 2 * TILE_F * sizeof(float);

    RetrievalLoss_26946624815701_kernel<<<blocks, BLOCK_THREADS, shmem, stream>>>(
        q, dm, out, nrows);
}